// LocalSlidingWindowAttention_64201171141503
// MI455X (gfx1250) — compile-verified
//
#include <hip/hip_runtime.h>
#include <hip/hip_bf16.h>

// Sliding-window attention (win=64, W=129) for B=8, N=8192, E=128, H=8, D=16.
// All matrix math uses V_WMMA_F32_16X16X4_F32 (full f32, matching reference;
// the workload is memory-bound on the 270MB probs write, so fp32 WMMA is the
// right precision/path on MI455X).

typedef float v2f __attribute__((ext_vector_type(2)));
typedef float v8f __attribute__((ext_vector_type(8)));

namespace {
constexpr int Bn = 8;
constexpr int Nn = 8192;
constexpr int En = 128;
constexpr int Hn = 8;
constexpr int Dn = 16;
constexpr int WIN = 64;
constexpr int WW = 2 * WIN + 1;   // 129
constexpr int KT = 9;             // key tiles per 16-query tile (144 keys)
}

__device__ __forceinline__ v8f wmma4(v2f a, v2f b, v8f c) {
  // 8 args: (neg_a, A, neg_b, B, c_mod, C, reuse_a, reuse_b)
  return __builtin_amdgcn_wmma_f32_16x16x4_f32(false, a, false, b, (short)0, c,
                                               false, false);
}

__device__ __forceinline__ v8f zero8() {
  v8f z = {0.f, 0.f, 0.f, 0.f, 0.f, 0.f, 0.f, 0.f};
  return z;
}

// ---------------------------------------------------------------------------
// Kernel 1: projection  out[b,h,n,d] = sum_k x[b,n,k]*W[h*16+d, k] + bias
// grid.x = B*N/16, block = 256 (8 waves; wave w handles head w's 16 features)
// ---------------------------------------------------------------------------
__global__ __launch_bounds__(256) void qkv_proj_kernel(
    const float* __restrict__ x, const float* __restrict__ W,
    const float* __restrict__ bias, float* __restrict__ out) {
  const int tid = threadIdx.x;
  const int lane = tid & 31;
  const int wave = tid >> 5;
  const int row16 = lane & 15;          // M (A) / N (B,C) index
  const int kgrp = (lane >> 4) << 1;    // 0 or 2: K pair selected by half-wave
  const int rofs = (lane >> 4) << 3;    // C/D row offset for upper half-wave
  const int m0 = blockIdx.x * 16;       // token tile
  const int e0 = wave * 16;             // output feature tile (== head*16)

  v8f c = zero8();
  const float* arow = x + (size_t)(m0 + row16) * En;
  const float* brow = W + (size_t)(e0 + row16) * En;
#pragma unroll 4
  for (int k0 = 0; k0 < En; k0 += 4) {
    v2f a = *(const v2f*)(arow + k0 + kgrp);
    v2f bf = *(const v2f*)(brow + k0 + kgrp);
    c = wmma4(a, bf, c);
  }
  const float bb = bias[e0 + row16];

  const int bi = m0 / Nn;
  const int nl = m0 % Nn;
  const size_t obase = (((size_t)bi * Hn + wave) * Nn) * Dn;
#pragma unroll
  for (int r = 0; r < 8; ++r) {
    const int n = nl + r + rofs;
    out[obase + (size_t)n * Dn + row16] = c[r] + bb;
  }
}

// ---------------------------------------------------------------------------
// Kernel 2: sliding-window attention per (16-query tile, head).
// grid = (N/16, B, 2); block = 128 (4 waves; wave w -> head z*4+w)
// ---------------------------------------------------------------------------
__global__ __launch_bounds__(128) void swa_kernel(
    const float* __restrict__ Q, const float* __restrict__ K,
    const float* __restrict__ V, float* __restrict__ probs,
    float* __restrict__ attn) {
  __shared__ float sP[4][16][148];  // 16 rows x 144 staged cols (+pad)

  const int tid = threadIdx.x;
  const int lane = tid & 31;
  const int wave = tid >> 5;
  const int row16 = lane & 15;
  const int kgrp = (lane >> 4) << 1;
  const int rofs = (lane >> 4) << 3;

  const int n0 = blockIdx.x * 16;
  const int bi = blockIdx.y;
  const int h = blockIdx.z * 4 + wave;
  const size_t base = (((size_t)bi * Hn + h) * Nn) * Dn;
  const float* Qh = Q + base;
  const float* Kh = K + base;
  const float* Vh = V + base;

  // Preload Q A-fragments (full D=16: 4 k-steps of 4)
  v2f qa[4];
#pragma unroll
  for (int kk = 0; kk < 4; ++kk)
    qa[kk] = *(const v2f*)(Qh + (size_t)(n0 + row16) * Dn + kk * 4 + kgrp);

  // ---- scores: S = (Q * K^T) / sqrt(D), staged to LDS -------------------
  for (int t = 0; t < KT; ++t) {
    const int key0 = n0 - WIN + t * 16;
    const int key = key0 + row16;            // B-fragment: N index = key
    const bool ok = (key >= 0) && (key < Nn);
    v8f acc = zero8();
#pragma unroll
    for (int kk = 0; kk < 4; ++kk) {
      v2f bf;
      if (ok) {
        bf = *(const v2f*)(Kh + (size_t)key * Dn + kk * 4 + kgrp);
      } else {
        bf.x = 0.f; bf.y = 0.f;              // zero-padded key -> score 0
      }
      acc = wmma4(qa[kk], bf, acc);
    }
#pragma unroll
    for (int r = 0; r < 8; ++r)
      sP[wave][r + rofs][t * 16 + row16] = acc[r] * 0.25f;  // 1/sqrt(16)
  }
  __syncthreads();

  // ---- softmax over the 129-entry window of each row --------------------
  // Row r (global query n0+r) uses staged columns [r, r+128].
  if (lane < 16) {
    const int r = lane;
    float* rowp = &sP[wave][r][0];
    float mx = -3.402823466e38f;
    for (int w = 0; w < WW; ++w) mx = fmaxf(mx, rowp[r + w]);
    float sum = 0.f;
    for (int w = 0; w < WW; ++w) sum += expf(rowp[r + w] - mx);
    const float inv = 1.0f / sum;
    for (int w = 0; w < WW; ++w) rowp[r + w] = expf(rowp[r + w] - mx) * inv;
    for (int c = 0; c < r; ++c) rowp[c] = 0.f;          // outside window: 0
    for (int c = r + WW; c < 144; ++c) rowp[c] = 0.f;
  }
  __syncthreads();

  // ---- write probs[b,h,n,w] (coalesced-ish over w) ----------------------
  const size_t pbase = (((size_t)bi * Hn + h) * Nn + n0) * (size_t)WW;
  for (int i = lane; i < 16 * WW; i += 32) {
    const int r = i / WW;
    const int w = i - r * WW;
    probs[pbase + (size_t)r * WW + w] = sP[wave][r][r + w];
  }

  // ---- O = P (16x144) * V (144x16) --------------------------------------
  v8f o = zero8();
  for (int t = 0; t < KT; ++t) {
    const int key0 = n0 - WIN + t * 16;
#pragma unroll
    for (int kk = 0; kk < 4; ++kk) {
      const int kof = kk * 4 + kgrp;
      v2f a;
      a.x = sP[wave][row16][t * 16 + kof];
      a.y = sP[wave][row16][t * 16 + kof + 1];
      v2f bf;
      const int k0i = key0 + kof;
      const int k1i = k0i + 1;
      bf.x = (k0i >= 0 && k0i < Nn) ? Vh[(size_t)k0i * Dn + row16] : 0.f;
      bf.y = (k1i >= 0 && k1i < Nn) ? Vh[(size_t)k1i * Dn + row16] : 0.f;
      o = wmma4(a, bf, o);
    }
  }
#pragma unroll
  for (int r = 0; r < 8; ++r)
    attn[base + (size_t)(n0 + r + rofs) * Dn + row16] = o[r];
}

// ---------------------------------------------------------------------------
// Kernel 3: out[b,n,:] = attn[b,:,n,:].flatten() @ Wo^T + bo
// grid.x = B*N/16, block = 256
// ---------------------------------------------------------------------------
__global__ __launch_bounds__(256) void out_proj_kernel(
    const float* __restrict__ attn, const float* __restrict__ Wo,
    const float* __restrict__ bo, float* __restrict__ out) {
  const int tid = threadIdx.x;
  const int lane = tid & 31;
  const int wave = tid >> 5;
  const int row16 = lane & 15;
  const int kgrp = (lane >> 4) << 1;
  const int rofs = (lane >> 4) << 3;
  const int m0 = blockIdx.x * 16;
  const int e0 = wave * 16;
  const int bi = m0 / Nn;
  const int nl = m0 % Nn;

  v8f c = zero8();
  const float* brow = Wo + (size_t)(e0 + row16) * En;
#pragma unroll 4
  for (int k0 = 0; k0 < En; k0 += 4) {
    const int k = k0 + kgrp;                 // pair (k,k+1) stays in one head
    const int hh = k >> 4;
    const int d = k & 15;
    v2f a = *(const v2f*)(attn + (((size_t)bi * Hn + hh) * Nn + nl + row16) * Dn + d);
    v2f bf = *(const v2f*)(brow + k);
    c = wmma4(a, bf, c);
  }
  const float bb = bo[e0 + row16];
#pragma unroll
  for (int r = 0; r < 8; ++r)
    out[(size_t)(m0 + r + rofs) * En + e0 + row16] = c[r] + bb;
}

// ---------------------------------------------------------------------------
extern "C" void kernel_launch(void* const* d_in, const int* in_sizes, int n_in,
                              void* d_out, int out_size, void* d_ws,
                              size_t ws_size, hipStream_t stream) {
  const float* q  = (const float*)d_in[0];
  const float* Wq = (const float*)d_in[1];
  const float* bq = (const float*)d_in[2];
  const float* Wk = (const float*)d_in[3];
  const float* bk = (const float*)d_in[4];
  const float* Wv = (const float*)d_in[5];
  const float* bv = (const float*)d_in[6];
  const float* Wo = (const float*)d_in[7];
  const float* bo = (const float*)d_in[8];

  float* out = (float*)d_out;                          // [B,N,E]
  float* probs = out + (size_t)Bn * Nn * En;           // [B,H,N,W]

  const size_t SEG = (size_t)Bn * Hn * Nn * Dn;        // 8.39M floats
  float* Qb = (float*)d_ws;
  float* Kb = Qb + SEG;
  float* Vb = Kb + SEG;
  float* Ab = Vb + SEG;                                // attention out [B,H,N,D]

  const dim3 gProj(Bn * Nn / 16);
  const dim3 bProj(256);
  qkv_proj_kernel<<<gProj, bProj, 0, stream>>>(q, Wq, bq, Qb);
  qkv_proj_kernel<<<gProj, bProj, 0, stream>>>(q, Wk, bk, Kb);
  qkv_proj_kernel<<<gProj, bProj, 0, stream>>>(q, Wv, bv, Vb);

  swa_kernel<<<dim3(Nn / 16, Bn, 2), dim3(128), 0, stream>>>(Qb, Kb, Vb, probs, Ab);

  out_proj_kernel<<<gProj, bProj, 0, stream>>>(Ab, Wo, bo, out);
}